// SphereTracingRenderer_11020886082109
// MI455X (gfx1250) — compile-verified
//
#include <hip/hip_runtime.h>

typedef float v2f __attribute__((ext_vector_type(2)));
typedef float v8f __attribute__((ext_vector_type(8)));

#define BLOCK 256
#define MAX_ITERS 32
#define EPS_HIT 5e-7f

__device__ __forceinline__ float lane_gather(float v, int srcLane) {
    // ds_bpermute: dst[lane] = src[index[lane]]; index in bytes
    return __int_as_float(__builtin_amdgcn_ds_bpermute(srcLane << 2, __float_as_int(v)));
}

// raw v_sqrt_f32 (~1 ulp) -- avoids the IEEE Newton/fixup tail on the
// serially-dependent trace chain
__device__ __forceinline__ float fast_sqrtf(float x) {
    return __builtin_amdgcn_sqrtf(x);
}
// raw v_rcp_f32 -- avoids div_scale/fixup sequence in the sigmoid
__device__ __forceinline__ float fast_rcpf(float x) {
    return __builtin_amdgcn_rcpf(x);
}

__global__ __launch_bounds__(BLOCK)
void SphereTracingRenderer_kernel(const float* __restrict__ origins,
                                  const float* __restrict__ dirs,
                                  const float* __restrict__ center,
                                  const float* __restrict__ radius,
                                  const float* __restrict__ Wc,
                                  const float* __restrict__ bc,
                                  float* __restrict__ out,
                                  int nrays)
{
    // per-wave LDS scratch: 32 rays x 4 floats (3 colors + pad)
    __shared__ float lds[(BLOCK / 32) * 32 * 4];

    const int tid  = threadIdx.x;
    const int lane = tid & 31;
    const int wave = tid >> 5;
    int gid = blockIdx.x * BLOCK + tid;
    // keep EXEC all-1s for WMMA: clamp instead of early-exit (no tail for
    // N=2M anyway; duplicate work is benign/identical)
    if (gid >= nrays) gid = nrays - 1;

    const float cx = center[0], cy = center[1], cz = center[2];
    const float r  = radius[0];

    const float ox = origins[3 * gid + 0];
    const float oy = origins[3 * gid + 1];
    const float oz = origins[3 * gid + 2];
    const float dx = dirs[3 * gid + 0];
    const float dy = dirs[3 * gid + 1];
    const float dz = dirs[3 * gid + 2];

    const float ocx = ox - cx, ocy = oy - cy, ocz = oz - cz;

    // ---- sphere-trace loop: z_{i+1} = z_i + (||d*z_i + (o-c)|| - r) ----
    float z = 1.0f;
#pragma unroll 4
    for (int it = 0; it < MAX_ITERS - 1; ++it) {
        float qx = fmaf(dx, z, ocx);
        float qy = fmaf(dy, z, ocy);
        float qz = fmaf(dz, z, ocz);
        float d2 = fmaf(qx, qx, fmaf(qy, qy, qz * qz));
        z += fast_sqrtf(d2) - r;
    }

    // final points + hit mask
    const float px = fmaf(dx, z, ox);
    const float py = fmaf(dy, z, oy);
    const float pz = fmaf(dz, z, oz);
    const float qx = px - cx, qy = py - cy, qz = pz - cz;
    const float sd = fast_sqrtf(fmaf(qx, qx, fmaf(qy, qy, qz * qz))) - r;
    const float maskf = (sd <= EPS_HIT) ? 1.0f : 0.0f;

    // ---- color head via V_WMMA_F32_16X16X4_F32 ----
    // A (16x4) = [px py pz 1] rows; B (4x16) = [Wc row0; row1; row2; bc]
    // so D = A*B computes p@Wc + bc for 16 rays in one WMMA.
    //
    // B layout (32-bit B, 4x16): VGPR0: lanes0-15 = K0, lanes16-31 = K2;
    //                            VGPR1: lanes0-15 = K1, lanes16-31 = K3
    const int n = lane & 15;
    v2f B; B.x = 0.0f; B.y = 0.0f;
    if (n < 3) {
        if (lane < 16) { B.x = Wc[0 * 3 + n]; B.y = Wc[1 * 3 + n]; }
        else           { B.x = Wc[2 * 3 + n]; B.y = bc[n];         }
    }

    // A layout (32-bit A, 16x4): lanes0-15 hold (K0,K1) of row M=lane;
    //                            lanes16-31 hold (K2,K3) of row M=lane-16
    const float z_low  = lane_gather(pz, lane & 15);        // pz of low-half partner
    const float x_high = lane_gather(px, (lane & 15) + 16); // px of high-half partner
    const float y_high = lane_gather(py, (lane & 15) + 16);

    v2f A0, A1;
    if (lane < 16) { A0.x = px;    A0.y = py;   A1.x = x_high; A1.y = y_high; }
    else           { A0.x = z_low; A0.y = 1.0f; A1.x = pz;     A1.y = 1.0f;   }

    v8f acc0 = {};
    v8f acc1 = {};
    acc0 = __builtin_amdgcn_wmma_f32_16x16x4_f32(false, A0, false, B,
                                                 (short)0, acc0, false, false);
    acc1 = __builtin_amdgcn_wmma_f32_16x16x4_f32(false, A1, false, B,
                                                 (short)0, acc1, false, false);

    // ---- scatter D (16x16 f32: VGPR v holds row M=v (lanes0-15) / M=v+8
    // (lanes16-31), N = lane%16) into LDS [ray-in-wave][4], cols 0..2 ----
    float* wlds = &lds[wave * 32 * 4];
    if (n < 3) {
        const int rbase = (lane < 16) ? 0 : 8;
#pragma unroll
        for (int v = 0; v < 8; ++v) {
            wlds[(rbase + v) * 4 + n]      = acc0[v];   // rays 0..15 of wave
            wlds[(16 + rbase + v) * 4 + n] = acc1[v];   // rays 16..31 of wave
        }
    }
    __builtin_amdgcn_wave_barrier();   // wave-synchronous LDS exchange

    const float h0 = wlds[lane * 4 + 0];
    const float h1 = wlds[lane * 4 + 1];
    const float h2 = wlds[lane * 4 + 2];

    // masked sigmoid: maskf * 1/(1+exp(-h))
    const float c0 = maskf * fast_rcpf(1.0f + __expf(-h0));
    const float c1 = maskf * fast_rcpf(1.0f + __expf(-h1));
    const float c2 = maskf * fast_rcpf(1.0f + __expf(-h2));

    out[3 * gid + 0] = c0;
    out[3 * gid + 1] = c1;
    out[3 * gid + 2] = c2;
}

extern "C" void kernel_launch(void* const* d_in, const int* in_sizes, int n_in,
                              void* d_out, int out_size, void* d_ws, size_t ws_size,
                              hipStream_t stream) {
    const float* origins = (const float*)d_in[0];
    const float* dirs    = (const float*)d_in[1];
    const float* center  = (const float*)d_in[2];
    const float* radius  = (const float*)d_in[3];
    const float* Wc      = (const float*)d_in[4];
    const float* bc      = (const float*)d_in[5];
    float* out = (float*)d_out;

    const int nrays  = in_sizes[0] / 3;          // 2,097,152
    const int blocks = (nrays + BLOCK - 1) / BLOCK;

    SphereTracingRenderer_kernel<<<blocks, BLOCK, 0, stream>>>(
        origins, dirs, center, radius, Wc, bc, out, nrays);
}